// Mutual_Informations_23605140259219
// MI455X (gfx1250) — compile-verified
//
#include <hip/hip_runtime.h>

// Mutual_Informations for MI455X (gfx1250), wave32.
// f_p: [8,64,256,256] f32, f_ms: [8,64,16,16] f32.
// d_out = rel_p [8,64,256,256] ++ rel_ms [8,64,16,16].

typedef float v2f __attribute__((ext_vector_type(2)));
typedef float v8f __attribute__((ext_vector_type(8)));

// ---------- helpers ----------
__device__ __forceinline__ unsigned f2ord(float f) {
  unsigned u = __float_as_uint(f);
  return (u & 0x80000000u) ? ~u : (u | 0x80000000u);
}
__device__ __forceinline__ float ord2f(unsigned k) {
  unsigned u = (k & 0x80000000u) ? (k & 0x7FFFFFFFu) : ~k;
  return __uint_as_float(u);
}
// bilinear (half-pixel, 16x upsample) weight of fine index (with j0=floor,frac) onto coarse j
__device__ __forceinline__ float wgt(int j0, float fr, int j) {
  int a0 = j0 < 0 ? 0 : (j0 > 15 ? 15 : j0);
  int j1 = j0 + 1;
  int a1 = j1 < 0 ? 0 : (j1 > 15 ? 15 : j1);
  float w = 0.0f;
  if (a0 == j) w += 1.0f - fr;
  if (a1 == j) w += fr;
  return w;
}
__device__ __forceinline__ float bwf(int f, int c) {
  float x = ((float)f - 7.5f) * 0.0625f;
  float fl = floorf(x);
  return wgt((int)fl, x - fl, c);
}
__device__ __forceinline__ float entf(float cnt) {  // -p*log(p+eps), p = cnt/65536
  float p = cnt * (1.0f / 65536.0f);
  return -p * __logf(p + 1e-8f);
}
__device__ __forceinline__ float blkSum(float v, float* red, int n) {
  int t = threadIdx.x;
  red[t] = v; __syncthreads();
  for (int o = n >> 1; o > 0; o >>= 1) { if (t < o) red[t] += red[t + o]; __syncthreads(); }
  float r = red[0]; __syncthreads();
  return r;
}
__device__ __forceinline__ float blkMin(float v, float* red, int n) {
  int t = threadIdx.x;
  red[t] = v; __syncthreads();
  for (int o = n >> 1; o > 0; o >>= 1) { if (t < o) red[t] = fminf(red[t], red[t + o]); __syncthreads(); }
  float r = red[0]; __syncthreads();
  return r;
}
__device__ __forceinline__ float blkMax(float v, float* red, int n) {
  int t = threadIdx.x;
  red[t] = v; __syncthreads();
  for (int o = n >> 1; o > 0; o >>= 1) { if (t < o) red[t] = fmaxf(red[t], red[t + o]); __syncthreads(); }
  float r = red[0]; __syncthreads();
  return r;
}

// ---------- K0: init min/max slots + K = u^T u (16x16) ----------
__global__ __launch_bounds__(256) void k_init(unsigned* __restrict__ minmax,
                                              float* __restrict__ Kmat) {
  int t = threadIdx.x;
  if (t < 16) minmax[t] = (t & 1) ? 0u : 0xFFFFFFFFu;  // [b*2]=min, [b*2+1]=max
  int i = t >> 4, i2 = t & 15;
  float s = 0.0f;
  for (int f = 0; f < 256; ++f) s += bwf(f, i) * bwf(f, i2);
  Kmat[t] = s;
}

// ---------- K1: one pass over f_p per (b,c): pooled 16x16 mean, G = U^T f_p, sum/sumsq ----------
__global__ __launch_bounds__(256) void k_pass1(const float* __restrict__ fp,
    float* __restrict__ pooled, float* __restrict__ G,
    float* __restrict__ bm, float* __restrict__ nbv) {
  int bc = blockIdx.x, t = threadIdx.x;
  const float* src = fp + (size_t)bc * 65536;
  __shared__ float row[256];
  __shared__ float Hrow[256][16];
  __shared__ float Brow[256][16];
  __shared__ float red[256];
  __shared__ int   pj0[256];
  __shared__ float pfr[256];
  {
    float x = ((float)t - 7.5f) * 0.0625f;
    float fl = floorf(x);
    pj0[t] = (int)fl;
    pfr[t] = x - fl;
  }
  float s = 0.0f, s2 = 0.0f;
  __syncthreads();
  for (int h = 0; h < 256; ++h) {
    float v = src[h * 256 + t];      // coalesced
    row[t] = v;
    s += v; s2 += v * v;
    __syncthreads();
    if (t < 16) {
      int j = t;
      float hs = 0.0f, bs = 0.0f;
      int wlo = 16 * j - 8;  if (wlo < 0) wlo = 0;
      int whi = 16 * j + 23; if (whi > 255) whi = 255;
      for (int w = wlo; w <= whi; ++w) hs += wgt(pj0[w], pfr[w], j) * row[w];
      for (int w = 16 * j; w < 16 * j + 16; ++w) bs += row[w];
      Hrow[h][j] = hs;
      Brow[h][j] = bs;
    }
    __syncthreads();
  }
  int i = t >> 4, j = t & 15;
  float ps = 0.0f;
  for (int h = 16 * i; h < 16 * i + 16; ++h) ps += Brow[h][j];
  pooled[(size_t)bc * 256 + t] = ps * (1.0f / 256.0f);
  float gs = 0.0f;
  int hlo = 16 * i - 8;  if (hlo < 0) hlo = 0;
  int hhi = 16 * i + 23; if (hhi > 255) hhi = 255;
  for (int h = hlo; h <= hhi; ++h) gs += wgt(pj0[h], pfr[h], i) * Hrow[h][j];
  G[(size_t)bc * 256 + t] = gs;
  float sumAll = blkSum(s, red, 256);
  float sqAll  = blkSum(s2, red, 256);
  if (t == 0) {
    float m = sumAll * (1.0f / 65536.0f);
    bm[bc] = m;
    nbv[bc] = sqrtf(fmaxf(sqAll - 65536.0f * m * m, 0.0f));
  }
}

// ---------- K2: channel-mean image f_p_m + per-b min/max ----------
__global__ __launch_bounds__(256) void k_chanmean(const float* __restrict__ fp,
    float* __restrict__ fpm, unsigned* __restrict__ minmax) {
  int b = blockIdx.y, t = threadIdx.x;
  int pix = blockIdx.x * 256 + t;
  const float* base = fp + (size_t)b * 64 * 65536 + pix;
  float s = 0.0f;
  for (int c = 0; c < 64; ++c) s += base[(size_t)c * 65536];
  float m = s * (1.0f / 64.0f);
  fpm[(size_t)b * 65536 + pix] = m;
  __shared__ float red[256];
  float mn = blkMin(m, red, 256);
  float mx = blkMax(m, red, 256);
  if (t == 0) {
    atomicMin(&minmax[b * 2 + 0], f2ord(mn));
    atomicMax(&minmax[b * 2 + 1], f2ord(mx));
  }
}

// ---------- K3: histogram + entropy of normalized f_p_m -> h_p[b] ----------
__global__ __launch_bounds__(256) void k_hp(const float* __restrict__ fpm,
    const unsigned* __restrict__ minmax, float* __restrict__ hp) {
  int b = blockIdx.x, t = threadIdx.x;
  __shared__ int hist[256];
  __shared__ float red[256];
  hist[t] = 0; __syncthreads();
  float mn = ord2f(minmax[b * 2 + 0]);
  float mx = ord2f(minmax[b * 2 + 1]);
  float sc = 255.0f / (mx - mn);
  const float* src = fpm + (size_t)b * 65536;
  for (int k = t; k < 65536; k += 256) {
    int bin = (int)((src[k] - mn) * sc);
    bin = bin < 0 ? 0 : (bin > 255 ? 255 : bin);
    atomicAdd(&hist[bin], 1);
  }
  __syncthreads();
  float p = (float)hist[t] * (1.0f / 65536.0f);
  float h = blkSum(-p * __logf(p + 1e-8f), red, 256);
  if (t == 0) hp[b] = h;
}

// ---------- K4: per (b,c) f_ms: bins, histogram, entropy, mean am, na via quad form ----------
__global__ __launch_bounds__(256) void k_ms(const float* __restrict__ fms,
    const float* __restrict__ Kmat, int* __restrict__ msbin, int* __restrict__ hist_ms,
    float* __restrict__ hms, float* __restrict__ amv, float* __restrict__ nav) {
  int bc = blockIdx.x, t = threadIdx.x;
  __shared__ float v[256], red[256], T1[256], T2[256], Ks[256];
  __shared__ int hist[256];
  float x = fms[(size_t)bc * 256 + t];
  v[t] = x; Ks[t] = Kmat[t]; hist[t] = 0;
  __syncthreads();
  float mn = blkMin(x, red, 256);
  float mx = blkMax(x, red, 256);
  int bin = (int)((x - mn) * (255.0f / (mx - mn)));
  bin = bin < 0 ? 0 : (bin > 255 ? 255 : bin);
  msbin[(size_t)bc * 256 + t] = bin;
  atomicAdd(&hist[bin], 1);
  float am = blkSum(x, red, 256) * (1.0f / 256.0f);
  // quad = ms . (K ms K^T) = sum_p ms_u^2
  int i = t >> 4, j2 = t & 15;
  float a1 = 0.0f;
  for (int jj = 0; jj < 16; ++jj) a1 += v[i * 16 + jj] * Ks[jj * 16 + j2];
  T1[t] = a1;
  __syncthreads();
  float a2 = 0.0f;
  for (int ii = 0; ii < 16; ++ii) a2 += Ks[ii * 16 + i] * T1[ii * 16 + j2];
  T2[t] = a2;
  float quad = blkSum(v[t] * T2[t], red, 256);
  float p = (float)hist[t] * (1.0f / 256.0f);
  float h = blkSum(-p * __logf(p + 1e-8f), red, 256);
  hist_ms[(size_t)bc * 256 + t] = hist[t];
  if (t == 0) {
    hms[bc] = h;
    amv[bc] = am;
    nav[bc] = sqrtf(fmaxf(quad - 65536.0f * am * am, 0.0f));
  }
}

// ---------- K5: f_p_a = channel-mean of pooled; bins + histogram ----------
__global__ __launch_bounds__(256) void k_pa(const float* __restrict__ pooled,
    int* __restrict__ pabin, int* __restrict__ hist_pa) {
  int b = blockIdx.x, t = threadIdx.x;
  __shared__ float red[256];
  __shared__ int hist[256];
  hist[t] = 0;
  float s = 0.0f;
  for (int c = 0; c < 64; ++c) s += pooled[((size_t)b * 64 + c) * 256 + t];
  float m = s * (1.0f / 64.0f);
  __syncthreads();
  float mn = blkMin(m, red, 256);
  float mx = blkMax(m, red, 256);
  int bin = (int)((m - mn) * (255.0f / (mx - mn)));
  bin = bin < 0 ? 0 : (bin > 255 ? 255 : bin);
  pabin[(size_t)b * 256 + t] = bin;
  atomicAdd(&hist[bin], 1);
  __syncthreads();
  hist_pa[(size_t)b * 256 + t] = hist[t];
}

// ---------- K6: joint entropy, dense base + sparse correction ----------
__global__ __launch_bounds__(256) void k_joint(const int* __restrict__ msbin,
    const int* __restrict__ hist_ms, const int* __restrict__ pabin,
    const int* __restrict__ hist_pa, float* __restrict__ hj) {
  int bc = blockIdx.x, b = bc >> 6, t = threadIdx.x;
  __shared__ int hm[256], hp[256], mb[256], pb[256];
  __shared__ float red[256];
  hm[t] = hist_ms[(size_t)bc * 256 + t];
  hp[t] = hist_pa[(size_t)b * 256 + t];
  mb[t] = msbin[(size_t)bc * 256 + t];
  pb[t] = pabin[(size_t)b * 256 + t];
  __syncthreads();
  // base: sum over 256x256 of f(256 - hm[i] - hp[j]) assuming J=0
  float acc = 0.0f;
  int hmi = hm[t];
  for (int jj = 0; jj < 256; ++jj) acc += entf((float)(256 - hmi - hp[jj]));
  // correction: per pixel pair (ki,kj) with multiplicity J, add (f(c0+2J)-f(c0)) once
  int ki = mb[t], kj = pb[t];
  int J = 0;
  for (int p = 0; p < 256; ++p) J += (mb[p] == ki && pb[p] == kj) ? 1 : 0;
  float c0 = (float)(256 - hm[ki] - hp[kj]);
  acc += (entf(c0 + 2.0f * (float)J) - entf(c0)) / (float)J;
  float h = blkSum(acc, red, 256);
  if (t == 0) hj[bc] = h;
}

// ---------- K7: mi = h_p + h_ms - h_joint; softmax over C; rel_ms ----------
__global__ __launch_bounds__(256) void k_relms(const float* __restrict__ fms,
    const float* __restrict__ hp, const float* __restrict__ hms,
    const float* __restrict__ hj, float* __restrict__ out_relms) {
  int b = blockIdx.x, t = threadIdx.x;
  __shared__ float mi[64];
  if (t < 64) mi[t] = hp[b] + hms[(size_t)b * 64 + t] - hj[(size_t)b * 64 + t];
  __syncthreads();
  if (t == 0) {
    float m = mi[0];
    for (int c = 1; c < 64; ++c) m = fmaxf(m, mi[c]);
    float s = 0.0f;
    for (int c = 0; c < 64; ++c) { mi[c] = __expf(mi[c] - m); s += mi[c]; }
    float inv = 1.0f / s;
    for (int c = 0; c < 64; ++c) mi[c] *= inv;
  }
  __syncthreads();
  for (int k = t; k < 64 * 256; k += 256) {
    int c = k >> 8;
    float v = fms[(size_t)b * 64 * 256 + k];
    out_relms[(size_t)b * 64 * 256 + k] = v * (1.0f + mi[c]);
  }
}

// ---------- K8: s_m via f32 WMMA: dot = MS(64x256) x G^T(256x64), then correction ----------
__global__ __launch_bounds__(32) void k_corr_wmma(const float* __restrict__ fms,
    const float* __restrict__ G, const float* __restrict__ amv,
    const float* __restrict__ bm, const float* __restrict__ nav,
    const float* __restrict__ nbv, float* __restrict__ sm) {
  int b = blockIdx.y;
  int tm = blockIdx.x >> 2, tn = blockIdx.x & 3;
  int lane = threadIdx.x;
  int li = lane & 15, hi = lane >> 4;
  const float* Ar = fms + ((size_t)b * 64 + tm * 16 + li) * 256;  // A row M=li
  const float* Br = G   + ((size_t)b * 64 + tn * 16 + li) * 256;  // B col N=li (B = G^T)
  v8f acc = {0.f, 0.f, 0.f, 0.f, 0.f, 0.f, 0.f, 0.f};
  for (int k = 0; k < 256; k += 4) {
    // A 16x4 f32: lane(M + 16*(K>=2)), vgpr0=K even, vgpr1=K odd
    v2f a = *(const v2f*)(Ar + k + 2 * hi);
    v2f bvec = *(const v2f*)(Br + k + 2 * hi);
    acc = __builtin_amdgcn_wmma_f32_16x16x4_f32(false, a, false, bvec,
                                                (short)0, acc, false, false);
  }
  // D 16x16 f32: lane L -> N = L%16, rows M = r + 8*(L>=16)
  int col = tn * 16 + li;
  float bmd = bm[(size_t)b * 64 + col];
  for (int r = 0; r < 8; ++r) {
    int rowc = tm * 16 + r + 8 * hi;
    float amc = amv[(size_t)b * 64 + rowc];
    float denom = nav[(size_t)b * 64 + rowc] * nbv[(size_t)b * 64 + rowc];  // row-indexed quirk
    float val = (acc[r] - 65536.0f * amc * bmd) * (100.0f / denom);
    sm[((size_t)b * 64 + rowc) * 64 + col] = val;
  }
}

// ---------- K9a: per-b argmax/softmax/grouping ----------
__global__ __launch_bounds__(64) void k_spatial_a(const float* __restrict__ sm,
    float* __restrict__ garr, int* __restrict__ uniq) {
  int b = blockIdx.x, c = threadIdx.x;
  __shared__ float red[64], g[64];
  __shared__ int cnt[64];
  const float* rowp = sm + ((size_t)b * 64 + c) * 64;
  float m = rowp[0]; int mi = 0;
  for (int d = 1; d < 64; ++d) { float v = rowp[d]; if (v > m) { m = v; mi = d; } }
  g[c] = 0.0f; cnt[c] = 0;
  __syncthreads();
  float mm = blkMax(m, red, 64);
  float e = __expf(m - mm);
  float se = blkSum(e, red, 64);
  float score = e / se;
  atomicAdd(&g[mi], score);
  atomicAdd(&cnt[mi], 1);
  __syncthreads();
  garr[(size_t)b * 64 + c] = g[c];
  float u = blkSum(cnt[c] > 0 ? 1.0f : 0.0f, red, 64);
  if (c == 0) uniq[b] = (int)(u + 0.5f);
}

// ---------- K9b: min_k across batch, stable descending sort, selected softmax ----------
__global__ __launch_bounds__(64) void k_spatial_b(const float* __restrict__ garr,
    const int* __restrict__ uniq, int* __restrict__ order, float* __restrict__ warr,
    int* __restrict__ minkp) {
  int t = threadIdx.x;
  __shared__ int kk;
  if (t == 0) {
    int k = uniq[0];
    for (int b = 1; b < 8; ++b) k = min(k, uniq[b]);
    kk = (k + 1) / 2;
    *minkp = kk;
  }
  __syncthreads();
  if (t < 8) {
    int b = t;
    float gl[64]; int used[64]; float sorted[64];
    for (int c = 0; c < 64; ++c) { gl[c] = garr[(size_t)b * 64 + c]; used[c] = 0; }
    for (int pos = 0; pos < 64; ++pos) {            // stable: first max wins ties
      float best = -1e30f; int bi = 0;
      for (int c = 0; c < 64; ++c)
        if (!used[c] && gl[c] > best) { best = gl[c]; bi = c; }
      used[bi] = 1;
      order[(size_t)b * 64 + pos] = bi;
      sorted[pos] = best;
    }
    int mk = kk;
    float m = sorted[0];
    for (int c = 1; c < mk; ++c) m = fmaxf(m, sorted[c]);
    float s = 0.0f; float ex[64];
    for (int c = 0; c < mk; ++c) { ex[c] = __expf(sorted[c] - m); s += ex[c]; }
    float inv = 1.0f / s;
    for (int c = 0; c < 64; ++c) warr[(size_t)b * 64 + c] = (c < mk) ? ex[c] * inv : 0.0f;
  }
}

// ---------- K10: rel_p = f_p * (1 + sum_c w[c]*sigmoid(f_p[order[c]])) ----------
__global__ __launch_bounds__(256) void k_relp(const float* __restrict__ fp,
    const int* __restrict__ order, const float* __restrict__ warr,
    const int* __restrict__ minkp, float* __restrict__ out) {
  int b = blockIdx.y;
  int pix = blockIdx.x * 256 + threadIdx.x;
  int mk = *minkp;
  const float* base = fp + (size_t)b * 64 * 65536;
  float mask = 0.0f;
  for (int c = 0; c < mk; ++c) {
    int oc = order[(size_t)b * 64 + c];
    float x = base[(size_t)oc * 65536 + pix];
    mask += warr[(size_t)b * 64 + c] / (1.0f + __expf(-x));
  }
  float m1 = 1.0f + mask;
  for (int d = 0; d < 64; ++d) {
    float v = base[(size_t)d * 65536 + pix];
    out[((size_t)b * 64 + d) * 65536 + pix] = v * m1;
  }
}

// ---------- host ----------
extern "C" void kernel_launch(void* const* d_in, const int* in_sizes, int n_in,
                              void* d_out, int out_size, void* d_ws, size_t ws_size,
                              hipStream_t stream) {
  (void)in_sizes; (void)n_in; (void)out_size; (void)ws_size;
  const float* f_p  = (const float*)d_in[0];
  const float* f_ms = (const float*)d_in[1];
  float* out    = (float*)d_out;
  float* rel_p  = out;
  float* rel_ms = out + (size_t)8 * 64 * 65536;

  char* ws = (char*)d_ws;
  size_t off = 0;
  auto alloc = [&](size_t bytes) -> void* {
    void* p = ws + off;
    off += (bytes + 255) & ~(size_t)255;
    return p;
  };
  float*    pooled  = (float*)alloc(131072 * 4);   // [8,64,256]
  float*    G       = (float*)alloc(131072 * 4);   // [8,64,256]
  float*    fpm     = (float*)alloc(524288 * 4);   // [8,65536]
  float*    bm      = (float*)alloc(512 * 4);      // f_p channel means
  float*    nbv     = (float*)alloc(512 * 4);      // f_p channel norms
  float*    amv     = (float*)alloc(512 * 4);      // ms_u means
  float*    nav     = (float*)alloc(512 * 4);      // ms_u norms
  float*    hms     = (float*)alloc(512 * 4);
  float*    hj      = (float*)alloc(512 * 4);
  float*    hp      = (float*)alloc(8 * 4);
  unsigned* minmax  = (unsigned*)alloc(16 * 4);
  float*    Kmat    = (float*)alloc(256 * 4);
  int*      hist_ms = (int*)alloc(131072 * 4);
  int*      msbin   = (int*)alloc(131072 * 4);
  int*      hist_pa = (int*)alloc(2048 * 4);
  int*      pabin   = (int*)alloc(2048 * 4);
  float*    smv     = (float*)alloc(32768 * 4);    // [8,64,64]
  float*    garr    = (float*)alloc(512 * 4);
  int*      uniq    = (int*)alloc(8 * 4);
  int*      order   = (int*)alloc(512 * 4);
  float*    warr    = (float*)alloc(512 * 4);
  int*      minkp   = (int*)alloc(4);

  k_init     <<<1,            256, 0, stream>>>(minmax, Kmat);
  k_pass1    <<<512,          256, 0, stream>>>(f_p, pooled, G, bm, nbv);
  k_chanmean <<<dim3(256, 8), 256, 0, stream>>>(f_p, fpm, minmax);
  k_hp       <<<8,            256, 0, stream>>>(fpm, minmax, hp);
  k_ms       <<<512,          256, 0, stream>>>(f_ms, Kmat, msbin, hist_ms, hms, amv, nav);
  k_pa       <<<8,            256, 0, stream>>>(pooled, pabin, hist_pa);
  k_joint    <<<512,          256, 0, stream>>>(msbin, hist_ms, pabin, hist_pa, hj);
  k_relms    <<<8,            256, 0, stream>>>(f_ms, hp, hms, hj, rel_ms);
  k_corr_wmma<<<dim3(16, 8),   32, 0, stream>>>(f_ms, G, amv, bm, nav, nbv, smv);
  k_spatial_a<<<8,             64, 0, stream>>>(smv, garr, uniq);
  k_spatial_b<<<1,             64, 0, stream>>>(garr, uniq, order, warr, minkp);
  k_relp     <<<dim3(256, 8), 256, 0, stream>>>(f_p, order, warr, minkp, rel_p);
}